// ContourRec_11759620456533
// MI455X (gfx1250) — compile-verified
//
#include <hip/hip_runtime.h>
#include <hip/hip_bf16.h>

// ---------------------------------------------------------------------------
// CDNA5 (gfx1250) contourlet DFB reconstruction.
// Separable 12-tap convs implemented as band-matrix matmuls with
// V_WMMA_F32_16X16X4_F32 (wave32 WMMA). All permutation steps fused into
// LDS fills / epilogues. fp32 throughout (matches reference precision).
// ---------------------------------------------------------------------------

typedef __attribute__((ext_vector_type(2))) float v2f;
typedef __attribute__((ext_vector_type(8))) float v8f;

__device__ __forceinline__ v8f wmma4(v2f a, v2f b, v8f c) {
    // D = A(16x4) * B(4x16) + C(16x16), fp32
    return __builtin_amdgcn_wmma_f32_16x16x4_f32(
        /*neg_a=*/false, a, /*neg_b=*/false, b,
        /*c_mod=*/(short)0, c, /*reuse_a=*/false, /*reuse_b=*/false);
}

// 12-tap DFB filter (antisymmetric), f[i], zero outside [0,12)
__device__ __forceinline__ float fcoef(int i) {
    float r = 0.0f;
    r = (i == 0)  ?  0.0144f : r;
    r = (i == 1)  ?  0.0272f : r;
    r = (i == 2)  ?  0.0526f : r;
    r = (i == 3)  ?  0.0972f : r;
    r = (i == 4)  ?  0.193f  : r;
    r = (i == 5)  ?  0.63f   : r;
    r = (i == 6)  ? -0.63f   : r;
    r = (i == 7)  ? -0.193f  : r;
    r = (i == 8)  ? -0.0972f : r;
    r = (i == 9)  ? -0.0526f : r;
    r = (i == 10) ? -0.0272f : r;
    r = (i == 11) ? -0.0144f : r;
    return r;
}

// '2c' quincunx gather: x[r,c] from stage-1 p0/p1 planes (128x128 each)
__device__ __forceinline__ float gather2c(const float* __restrict__ p0pl,
                                          const float* __restrict__ p1pl,
                                          int r, int c) {
    int cp = (c + r) & 255;
    int j  = cp >> 1;
    if ((cp & 1) == 0) return p0pl[((r - j) & 127) * 128 + j];
    return p1pl[((r - 1 - j) & 127) * 128 + j];
}

// ---------------------------------------------------------------------------
// Stage 1: periodic 128x128 planes.  out = alpha*add + beta*conv(src)
// conv: out[r,c] = sum_{u,v} f[u] f[v] src[(r+u-OFS)&127, (c+v-OFS)&127]
// ---------------------------------------------------------------------------
template <int OFS>
__device__ __forceinline__ void s1_plane(const float* __restrict__ cs,
                                         const float* __restrict__ ap,
                                         float* __restrict__ op,
                                         float alpha, float beta) {
    __shared__ float pl[128 * 136];      // padded: 2*136 mod 64 == 16 -> disjoint banks
    __shared__ float scr[8][16 * 34];    // per-wave 16x32 Tmp, stride 34

    const int lane = threadIdx.x & 31;
    const int wave = threadIdx.x >> 5;
    const int m  = lane & 15;
    const int hi = lane >> 4;

    // Band-matrix constants: serve as A (pass1, F[r,k]=f[k-r]) and as
    // B (pass2, G[k,c]=f[k-c]) — identical per-lane values.
    v2f wF[7];
#pragma unroll
    for (int kk = 0; kk < 7; ++kk) {
        wF[kk].x = fcoef(4 * kk + 2 * hi - m);
        wF[kk].y = fcoef(4 * kk + 2 * hi + 1 - m);
    }

    // Fill plane into LDS (float4, coalesced)
#pragma unroll 4
    for (int i = 0; i < 16; ++i) {
        int v   = (int)threadIdx.x + i * 256;   // float4 index, 0..4095
        int row = v >> 5;                        // 32 float4 per row
        int c4  = v & 31;
        float4 d = ((const float4*)cs)[v];
        *(float4*)&pl[row * 136 + c4 * 4] = d;
    }
    __syncthreads();

    float* s = scr[wave];
    for (int t = wave; t < 64; t += 8) {   // 8x8 tiles of 16x16, uniform per wave
        const int R = (t >> 3) * 16;
        const int C = (t & 7) * 16;

        // Pass 1 (vertical): Tmp(16x32) = F(16x28) x In(28x32)
#pragma unroll
        for (int g = 0; g < 2; ++g) {
            v8f acc = {};
            const int cc = (C - OFS + 16 * g + m) & 127;
#pragma unroll
            for (int kk = 0; kk < 7; ++kk) {
                int r0 = R - OFS + 4 * kk + 2 * hi;
                v2f b;
                b.x = pl[(r0 & 127) * 136 + cc];
                b.y = pl[((r0 + 1) & 127) * 136 + cc];
                acc = wmma4(wF[kk], b, acc);
            }
            // D-layout -> scratch rows
#pragma unroll
            for (int i = 0; i < 8; ++i)
                s[(i + 8 * hi) * 34 + m + 16 * g] = acc[i];
        }
        asm volatile("" ::: "memory");   // LDS is in-order per wave

        // Pass 2 (horizontal): Out(16x16) = Tmp(16x28) x G(28x16)
        v8f acc = {};
#pragma unroll
        for (int kk = 0; kk < 7; ++kk) {
            v2f a = *(const v2f*)&s[m * 34 + 4 * kk + 2 * hi];
            acc = wmma4(a, wF[kk], acc);
        }
        asm volatile("" ::: "memory");

        // Epilogue: D layout == row segments -> coalesced stores
#pragma unroll
        for (int i = 0; i < 8; ++i) {
            int r = R + i + 8 * hi;
            int c = C + m;
            op[r * 128 + c] = alpha * ap[r * 128 + c] + beta * acc[i];
        }
    }
}

__global__ void __launch_bounds__(256)
k1_conv(const float* __restrict__ y0, const float* __restrict__ y1,
        const float* __restrict__ y2, const float* __restrict__ y3,
        float* __restrict__ p1s1) {
    const int p = blockIdx.x;           // 0..511, plane (b, CH) of the 64-ch stack
    const int b = p >> 6, CH = p & 63;
    const size_t half = (size_t)(b * 32 + (CH & 31)) * 16384;
    const float* cs = (CH < 32) ? (y1 + half) : (y2 + half);  // Y0
    const float* ap = (CH < 32) ? (y0 + half) : (y3 + half);  // Y1
    s1_plane<5>(cs, ap, p1s1 + (size_t)p * 16384,
                -0.70710678118654752f, -0.70710678118654752f);
}

__global__ void __launch_bounds__(256)
k2_conv(const float* __restrict__ y1, const float* __restrict__ y2,
        const float* __restrict__ p1s1, float* __restrict__ p0s1) {
    const int p = blockIdx.x;
    const int b = p >> 6, CH = p & 63;
    const size_t half = (size_t)(b * 32 + (CH & 31)) * 16384;
    const float* ap = (CH < 32) ? (y1 + half) : (y2 + half);  // Y0
    s1_plane<6>(p1s1 + (size_t)p * 16384, ap, p0s1 + (size_t)p * 16384,
                1.41421356237309505f, 1.0f);
}

// ---------------------------------------------------------------------------
// Stage 2: 128x256 planes, 'qper_col' extension (row wrap => col ^= 128).
// ---------------------------------------------------------------------------
__device__ __forceinline__ float s2_fetch(const float* __restrict__ pl,
                                          int rraw, int craw) {
    int c = craw & 255;
    int r = rraw;
    if (r < 0)        { r += 128; c ^= 128; }
    else if (r >= 128){ r -= 128; c ^= 128; }
    return pl[r * 264 + c];
}

template <int OFS, bool GATHER_FILL>
__device__ __forceinline__ void s2_plane(const float* __restrict__ fill_src,
                                         const float* __restrict__ f_p0,
                                         const float* __restrict__ f_p1,
                                         const float* __restrict__ a_p0,
                                         const float* __restrict__ a_p1,
                                         float* __restrict__ op,
                                         float alpha, float beta) {
    __shared__ float pl[128 * 264];      // 2*264 mod 64 == 16 -> disjoint banks
    __shared__ float scr[8][16 * 34];

    const int lane = threadIdx.x & 31;
    const int wave = threadIdx.x >> 5;
    const int m  = lane & 15;
    const int hi = lane >> 4;

    v2f wF[7];
#pragma unroll
    for (int kk = 0; kk < 7; ++kk) {
        wF[kk].x = fcoef(4 * kk + 2 * hi - m);
        wF[kk].y = fcoef(4 * kk + 2 * hi + 1 - m);
    }

    if constexpr (GATHER_FILL) {
        // Fill x0 plane via INVERSE '2c' map: coalesced reads, LDS scatter.
        // p0[rr,j] -> x[(rr+j)&127, (2j-r)&255]; p1[rr,j] -> r=(rr+1+j)&127.
#pragma unroll 4
        for (int i = 0; i < 64; ++i) {
            int v  = (int)threadIdx.x + i * 256;   // 0..16383
            int rr = v >> 7, j = v & 127;
            float a = f_p0[v];
            int r = (rr + j) & 127;
            int c = (2 * j - r) & 255;
            pl[r * 264 + c] = a;
            float bb = f_p1[v];
            r = (rr + 1 + j) & 127;
            c = (2 * j + 1 - r) & 255;
            pl[r * 264 + c] = bb;
        }
    } else {
#pragma unroll 4
        for (int i = 0; i < 32; ++i) {
            int v   = (int)threadIdx.x + i * 256;  // float4 index 0..8191
            int row = v >> 6;                       // 64 float4 per row
            int c4  = v & 63;
            float4 d = ((const float4*)fill_src)[v];
            *(float4*)&pl[row * 264 + c4 * 4] = d;
        }
    }
    __syncthreads();

    float* s = scr[wave];
    for (int t = wave; t < 128; t += 8) {  // 8x16 tiles, uniform per wave
        const int R = (t >> 4) * 16;
        const int C = (t & 15) * 16;

#pragma unroll
        for (int g = 0; g < 2; ++g) {
            v8f acc = {};
            const int cb = C - OFS + 16 * g + m;
#pragma unroll
            for (int kk = 0; kk < 7; ++kk) {
                int r0 = R - OFS + 4 * kk + 2 * hi;
                v2f b;
                b.x = s2_fetch(pl, r0, cb);
                b.y = s2_fetch(pl, r0 + 1, cb);
                acc = wmma4(wF[kk], b, acc);
            }
#pragma unroll
            for (int i = 0; i < 8; ++i)
                s[(i + 8 * hi) * 34 + m + 16 * g] = acc[i];
        }
        asm volatile("" ::: "memory");

        v8f acc = {};
#pragma unroll
        for (int kk = 0; kk < 7; ++kk) {
            v2f a = *(const v2f*)&s[m * 34 + 4 * kk + 2 * hi];
            acc = wmma4(a, wF[kk], acc);
        }
        asm volatile("" ::: "memory");

#pragma unroll
        for (int i = 0; i < 8; ++i) {
            int r = R + i + 8 * hi;
            int c = C + m;
            float addv = gather2c(a_p0, a_p1, r, c);  // L2-resident gather
            op[r * 256 + c] = alpha * addv + beta * acc[i];
        }
    }
}

__global__ void __launch_bounds__(256)
k4_conv(const float* __restrict__ p0s1, const float* __restrict__ p1s1,
        float* __restrict__ p1s2) {
    const int p = blockIdx.x;              // 0..255 = (b, ch)
    const int b = p >> 5, ch = p & 31;
    const size_t sA = (size_t)(b * 64 + ch) * 16384;        // x0 source
    const size_t sB = (size_t)(b * 64 + ch + 32) * 16384;   // x1 source
    s2_plane<5, true>(nullptr, p0s1 + sA, p1s1 + sA, p0s1 + sB, p1s1 + sB,
                      p1s2 + (size_t)p * 32768,
                      -0.70710678118654752f, -0.70710678118654752f);
}

__global__ void __launch_bounds__(256)
k5_conv(const float* __restrict__ p0s1, const float* __restrict__ p1s1,
        const float* __restrict__ p1s2, float* __restrict__ p0s2) {
    const int p = blockIdx.x;
    const int b = p >> 5, ch = p & 31;
    const size_t sA = (size_t)(b * 64 + ch) * 16384;        // x0 source
    s2_plane<6, false>(p1s2 + (size_t)p * 32768, nullptr, nullptr,
                       p0s1 + sA, p1s1 + sA,
                       p0s2 + (size_t)p * 32768,
                       1.41421356237309505f, 1.0f);
}

// ---------------------------------------------------------------------------
// Final '1r' re-interleave: out (8,32,256,256); coalesced float4 writes,
// scattered (L2-resident) reads of p0s2/p1s2 (each plane 128x256).
// ---------------------------------------------------------------------------
__global__ void __launch_bounds__(256)
k6_gather(const float* __restrict__ p0, const float* __restrict__ p1,
          float* __restrict__ out) {
    const int t = blockIdx.x * 256 + threadIdx.x;
    const size_t base = (size_t)t * 16;          // 16 floats per thread
    const int plane = (int)(base >> 16);         // 65536 per plane
    const int rem   = (int)(base & 65535);
    const int r = rem >> 8;
    const int c0 = rem & 255;
    const float* q0 = p0 + (size_t)plane * 32768;
    const float* q1 = p1 + (size_t)plane * 32768;

    float v[16];
#pragma unroll
    for (int k = 0; k < 16; ++k) {
        int c  = c0 + k;
        int rp = (r + c) & 255;
        int i  = rp >> 1;
        if ((rp & 1) == 0) v[k] = q0[i * 256 + ((c - i) & 255)];
        else               v[k] = q1[i * 256 + ((c - 1 - i) & 255)];
    }
    float4* o = (float4*)(out + base);
#pragma unroll
    for (int k = 0; k < 4; ++k)
        o[k] = *(const float4*)&v[4 * k];
}

// ---------------------------------------------------------------------------
extern "C" void kernel_launch(void* const* d_in, const int* in_sizes, int n_in,
                              void* d_out, int out_size, void* d_ws, size_t ws_size,
                              hipStream_t stream) {
    const float* y0 = (const float*)d_in[0];
    const float* y1 = (const float*)d_in[1];
    const float* y2 = (const float*)d_in[2];
    const float* y3 = (const float*)d_in[3];
    float* out = (float*)d_out;

    // Workspace layout: 4 x 32MB fp32 buffers (128 MB total)
    float* p1s1 = (float*)d_ws;              // (8,64,128,128)
    float* p0s1 = p1s1 + 8388608;            // (8,64,128,128)
    float* p1s2 = p0s1 + 8388608;            // (8,32,128,256)
    float* p0s2 = p1s2 + 8388608;            // (8,32,128,256)

    k1_conv<<<512, 256, 0, stream>>>(y0, y1, y2, y3, p1s1);
    k2_conv<<<512, 256, 0, stream>>>(y1, y2, p1s1, p0s1);
    k4_conv<<<256, 256, 0, stream>>>(p0s1, p1s1, p1s2);
    k5_conv<<<256, 256, 0, stream>>>(p0s1, p1s1, p1s2, p0s2);
    k6_gather<<<4096, 256, 0, stream>>>(p0s2, p1s2, out);
    (void)in_sizes; (void)n_in; (void)out_size; (void)ws_size;
}